// LightGCNEncoder_53437983097034
// MI455X (gfx1250) — compile-verified
//
#include <hip/hip_runtime.h>
#include <cstdint>

#define EMB 64
#define CHUNK 32                 // edges staged per wave per iteration
#define SPMM_BLOCKS 2048
#define SPMM_THREADS 256

// ---------------------------------------------------------------------------
// init: A = concat(user,item); acc(d_out) = same; B = 0      (float4 vectorized)
// ---------------------------------------------------------------------------
__global__ __launch_bounds__(256) void lgcn_init_kernel(
    const float4* __restrict__ u, const float4* __restrict__ it,
    float4* __restrict__ A, float4* __restrict__ acc, float4* __restrict__ B,
    int nU4, int n4)
{
    int i = blockIdx.x * blockDim.x + threadIdx.x;
    if (i >= n4) return;
    float4 v = (i < nU4) ? u[i] : it[i - nU4];
    A[i]   = v;
    acc[i] = v;
    B[i]   = make_float4(0.f, 0.f, 0.f, 0.f);
}

// acc += add; z = 0   (retire the old source buffer as the next destination)
__global__ __launch_bounds__(256) void lgcn_accum_zero_kernel(
    float4* __restrict__ acc, const float4* __restrict__ add,
    float4* __restrict__ z, int n4)
{
    int i = blockIdx.x * blockDim.x + threadIdx.x;
    if (i >= n4) return;
    float4 a = acc[i], b = add[i];
    a.x += b.x; a.y += b.y; a.z += b.z; a.w += b.w;
    acc[i] = a;
    z[i] = make_float4(0.f, 0.f, 0.f, 0.f);
}

// acc = (acc + add) * 0.25
__global__ __launch_bounds__(256) void lgcn_final_kernel(
    float4* __restrict__ acc, const float4* __restrict__ add, int n4)
{
    int i = blockIdx.x * blockDim.x + threadIdx.x;
    if (i >= n4) return;
    float4 a = acc[i], b = add[i];
    a.x = (a.x + b.x) * 0.25f;
    a.y = (a.y + b.y) * 0.25f;
    a.z = (a.z + b.z) * 0.25f;
    a.w = (a.w + b.w) * 0.25f;
    acc[i] = a;
}

// ---------------------------------------------------------------------------
// SpMM: dst[row,:] += val * src[col,:] for each COO edge.
//
// Each wave double-buffers 32-edge chunks of (rows,cols,vals) into its private
// LDS slice via global_load_async_to_lds_b32 (ASYNCcnt), pipelined with
// s_wait_asynccnt 3 so the next chunk's DMA overlaps current-chunk compute.
// Triples are stored interleaved in LDS: edge e occupies dwords [3e, 3e+2].
// Processing: the wave splits into two 16-lane halves; each half handles one
// edge, lane l4 covering floats [4*l4, 4*l4+3] of the 64-float row:
//   1x global_load_b128 gather from L2-resident src + 4x global_atomic_add_f32.
// Staged triples are read with plain (non-volatile) direct __shared__
// indexing so InferAddressSpaces emits ds_load_b32; the asm "memory"
// clobbers + ptrtoint capture of `stage` keep those reads ordered & live.
// ---------------------------------------------------------------------------
__global__ __launch_bounds__(SPMM_THREADS) void lgcn_spmm_kernel(
    const int*   __restrict__ rows,
    const int*   __restrict__ cols,
    const float* __restrict__ vals,
    const float* __restrict__ src,
    float*       __restrict__ dst,
    int nnz)
{
    __shared__ int stage[SPMM_THREADS / 32][2][3 * CHUNK];

    const int lane   = threadIdx.x & 31;
    const int wid    = threadIdx.x >> 5;
    const int gwave  = blockIdx.x * (blockDim.x >> 5) + wid;
    const int nwaves = gridDim.x * (blockDim.x >> 5);

    // Issue async staging of one 32-edge chunk into a wave-private LDS slice.
    // Lane i stages edge (b+i)'s triple interleaved at dwords [3i, 3i+2].
    auto issue = [&](int cb, long b) {
        // Low 32 bits of a generic pointer into LDS == LDS byte offset
        uint32_t off_r = (uint32_t)(uintptr_t)&stage[wid][cb][3 * lane];
        uint32_t off_c = (uint32_t)(uintptr_t)&stage[wid][cb][3 * lane + 1];
        uint32_t off_v = (uint32_t)(uintptr_t)&stage[wid][cb][3 * lane + 2];
        const int*   pr = rows + b + lane;
        const int*   pc = cols + b + lane;
        const float* pv = vals + b + lane;
        asm volatile("global_load_async_to_lds_b32 %0, %1, off"
                     :: "v"(off_r), "v"(pr) : "memory");
        asm volatile("global_load_async_to_lds_b32 %0, %1, off"
                     :: "v"(off_c), "v"(pc) : "memory");
        asm volatile("global_load_async_to_lds_b32 %0, %1, off"
                     :: "v"(off_v), "v"(pv) : "memory");
    };

    long base = (long)gwave * CHUNK;
    const long stride = (long)nwaves * CHUNK;
    if (base >= nnz) return;

    issue(0, base);
    int cb = 0;

    const int half = lane >> 4;      // which edge of the pair this lane works on
    const int l4   = lane & 15;      // float4 slot within the 64-float row

    while (base < nnz) {
        long nbase = base + stride;
        if (nbase < nnz) {
            issue(cb ^ 1, nbase);
            // Current chunk's 3 loads done; only next chunk's 3 outstanding.
            asm volatile("s_wait_asynccnt 0x3" ::: "memory");
        } else {
            asm volatile("s_wait_asynccnt 0x0" ::: "memory");
        }

        long remain = nnz - base;
        int count = remain < (long)CHUNK ? (int)remain : CHUNK;
        for (int e2 = 0; e2 < CHUNK / 2; ++e2) {
            int e = 2 * e2 + half;
            if (e < count) {
                int   r = stage[wid][cb][3 * e];          // ds_load_b32 (bcast)
                int   c = stage[wid][cb][3 * e + 1];
                float v = __int_as_float(stage[wid][cb][3 * e + 2]);

                const float4 x =
                    *(const float4*)(src + (size_t)c * EMB + 4 * l4);  // b128
                float* p = dst + (size_t)r * EMB + 4 * l4;
                unsafeAtomicAdd(p,     v * x.x);     // global_atomic_add_f32
                unsafeAtomicAdd(p + 1, v * x.y);
                unsafeAtomicAdd(p + 2, v * x.z);
                unsafeAtomicAdd(p + 3, v * x.w);
            }
        }

        cb ^= 1;
        base = nbase;
    }
}

// ---------------------------------------------------------------------------
extern "C" void kernel_launch(void* const* d_in, const int* in_sizes, int n_in,
                              void* d_out, int out_size, void* d_ws, size_t ws_size,
                              hipStream_t stream)
{
    const float* user_emb = (const float*)d_in[0];
    const float* item_emb = (const float*)d_in[1];
    const int*   adj_rows = (const int*)d_in[2];
    const int*   adj_cols = (const int*)d_in[3];
    const float* adj_vals = (const float*)d_in[4];

    const int numU = in_sizes[0] / EMB;       // 100000
    const int numI = in_sizes[1] / EMB;       // 50000
    const int N    = numU + numI;             // 150000
    const int nnz  = in_sizes[2];             // 4800000

    float* acc = (float*)d_out;               // accumulator == output layout
    float* A   = (float*)d_ws;                // ping buffer (N*64 floats)
    float* B   = A + (size_t)N * EMB;         // pong buffer (N*64 floats)

    const int n4  = N * (EMB / 4);            // float4 elements
    const int nU4 = numU * (EMB / 4);
    const int ew_blocks = (n4 + 255) / 256;

    // acc = A = concat(user, item); B = 0
    lgcn_init_kernel<<<ew_blocks, 256, 0, stream>>>(
        (const float4*)user_emb, (const float4*)item_emb,
        (float4*)A, (float4*)acc, (float4*)B, nU4, n4);

    // Layer 1: B += Adj * A ; acc += B ; zero A
    lgcn_spmm_kernel<<<SPMM_BLOCKS, SPMM_THREADS, 0, stream>>>(
        adj_rows, adj_cols, adj_vals, A, B, nnz);
    lgcn_accum_zero_kernel<<<ew_blocks, 256, 0, stream>>>(
        (float4*)acc, (const float4*)B, (float4*)A, n4);

    // Layer 2: A += Adj * B ; acc += A ; zero B
    lgcn_spmm_kernel<<<SPMM_BLOCKS, SPMM_THREADS, 0, stream>>>(
        adj_rows, adj_cols, adj_vals, B, A, nnz);
    lgcn_accum_zero_kernel<<<ew_blocks, 256, 0, stream>>>(
        (float4*)acc, (const float4*)A, (float4*)B, n4);

    // Layer 3: B += Adj * A ; acc = (acc + B) / 4
    lgcn_spmm_kernel<<<SPMM_BLOCKS, SPMM_THREADS, 0, stream>>>(
        adj_rows, adj_cols, adj_vals, A, B, nnz);
    lgcn_final_kernel<<<ew_blocks, 256, 0, stream>>>(
        (float4*)acc, (const float4*)B, n4);
}